// FeatureEmbeddingModule_37649683317258
// MI455X (gfx1250) — compile-verified
//
#include <hip/hip_runtime.h>
#include <cstdint>

typedef __attribute__((ext_vector_type(2))) float v2f;
typedef __attribute__((ext_vector_type(8))) float v8f;

#define BATCH 8
#define NPTS  4096
#define CPF   57
#define KNN   16
#define RAD2  0.04f
#define GPS   68   // padded LDS row stride (floats) to avoid bank conflicts

__device__ __forceinline__ v8f wmma4(v2f a, v2f b, v8f c) {
  // V_WMMA_F32_16X16X4_F32: D = A(16x4) * B(4x16) + C(16x16), all f32, wave32
  return __builtin_amdgcn_wmma_f32_16x16x4_f32(false, a, false, b, (short)0, c,
                                               false, false);
}

__device__ __forceinline__ float relu(float x) { return x > 0.0f ? x : 0.0f; }

// ---------------------------------------------------------------------------
// Tensor Data Mover: stage a contiguous f32 block (weights) into LDS.
// D# layout per CDNA5 ISA: count=1, type=2,
// data_size=4B, 1-row tile (tile_dim0 = nelem, tensor_dim1 = 1).
// ---------------------------------------------------------------------------
#if __has_builtin(__builtin_amdgcn_tensor_load_to_lds) && \
    __has_builtin(__builtin_amdgcn_s_wait_tensorcnt)
#define USE_TDM 1
typedef unsigned int v4u __attribute__((ext_vector_type(4)));
typedef int v4i __attribute__((ext_vector_type(4)));
typedef int v8it __attribute__((ext_vector_type(8)));

__device__ __forceinline__ void tdm_load_f32(const float* gsrc, float* ldst,
                                             unsigned int nelem) {
  unsigned long long ga = (unsigned long long)(uintptr_t)gsrc;
  // shared-aperture flat address keeps the LDS byte offset in addr[31:0]
  unsigned int laddr = (unsigned int)(uintptr_t)ldst;
  v4u g0;
  g0.x = 1u;                                                 // count = 1
  g0.y = laddr;                                              // lds_addr
  g0.z = (unsigned int)ga;                                   // global_addr lo
  g0.w = (unsigned int)((ga >> 32) & 0x01ffffffull) | (2u << 30);  // hi | type=2
  v8it g1;
  g1[0] = (int)(2u << 16);                              // data_size = 4 bytes
  g1[1] = (int)((nelem & 0xffffu) << 16);               // tensor_dim0[15:0]
  g1[2] = (int)(((nelem >> 16) & 0xffffu) | (1u << 16));// dim0 hi | tensor_dim1=1
  g1[3] = (int)((nelem & 0xffffu) << 16);               // tile_dim0 = nelem
  g1[4] = 0;                                            // tile_dim1/2 unused
  g1[5] = (int)nelem;                                   // tensor_dim0_stride
  g1[6] = 0;
  g1[7] = 0;
  v4i z4 = {0, 0, 0, 0};
#if defined(__clang_major__) && __clang_major__ >= 23
  v8it z8 = {0, 0, 0, 0, 0, 0, 0, 0};
  __builtin_amdgcn_tensor_load_to_lds(g0, g1, z4, z4, z8, 0);
#else
  __builtin_amdgcn_tensor_load_to_lds(g0, g1, z4, z4, 0);
#endif
}
#define TDM_WAIT() __builtin_amdgcn_s_wait_tensorcnt(0)
#else
#define USE_TDM 0
#define TDM_WAIT()
#endif

// ---------------------------------------------------------------------------
// Kernel 1: ball-KNN. 1 thread per query, candidates tiled through LDS.
// ---------------------------------------------------------------------------
__global__ __launch_bounds__(256) void knn_kernel(
    const float* __restrict__ xyz, int* __restrict__ idx_out,
    float* __restrict__ dist_out) {
  __shared__ float4 cand[256];
  const int b = blockIdx.x >> 4;              // NPTS/256 == 16 tiles per batch
  const int qtile = blockIdx.x & 15;
  const int q = qtile * 256 + threadIdx.x;
  const float* xb = xyz + (size_t)b * NPTS * 3;
  const float qx = xb[q * 3 + 0], qy = xb[q * 3 + 1], qz = xb[q * 3 + 2];
  const float qs = qx * qx + qy * qy + qz * qz;

  float best_d[KNN];
  int best_i[KNN];
#pragma unroll
  for (int j = 0; j < KNN; ++j) { best_d[j] = 3.0e38f; best_i[j] = q; }

  for (int c0 = 0; c0 < NPTS; c0 += 256) {
    __syncthreads();
    {
      const int c = c0 + threadIdx.x;
      const float x = xb[c * 3 + 0], y = xb[c * 3 + 1], z = xb[c * 3 + 2];
      cand[threadIdx.x] = make_float4(x, y, z, x * x + y * y + z * z);
    }
    __syncthreads();
    if (c0 + 256 < NPTS)  // global_prefetch_b8 of next candidate tile
      __builtin_prefetch(&xb[(c0 + 256 + threadIdx.x) * 3], 0, 3);
    for (int t = 0; t < 256; ++t) {
      const float4 cc = cand[t];
      float d = qs + cc.w - 2.0f * (qx * cc.x + qy * cc.y + qz * cc.z);
      d = fmaxf(d, 0.0f);
      if (d < best_d[KNN - 1]) {       // sorted ascending insertion
        float dj = d; int ij = c0 + t;
#pragma unroll
        for (int j = 0; j < KNN; ++j) {
          if (dj < best_d[j]) {
            float td = best_d[j]; int ti = best_i[j];
            best_d[j] = dj; best_i[j] = ij; dj = td; ij = ti;
          }
        }
      }
    }
  }
  const size_t base = ((size_t)b * NPTS + q) * KNN;
#pragma unroll
  for (int j = 0; j < KNN; ++j) {
    float d = best_d[j]; int i = best_i[j];
    if (d > RAD2) { d = 0.0f; i = q; }   // outside ball -> pad with self
    idx_out[base + j] = i;
    dist_out[base + j] = d;
  }
}

// ---------------------------------------------------------------------------
// Kernel 2: dynamic-filter MLP + gather + mean-over-K. 1 wave per point,
// the 16 neighbors are the WMMA M dimension. Weights staged by TDM.
// ---------------------------------------------------------------------------
__global__ __launch_bounds__(128) void dfg_kernel(
    const float* __restrict__ xyz, const float* __restrict__ points,
    const float* __restrict__ w1, const float* __restrict__ b1,
    const float* __restrict__ w2, const float* __restrict__ b2,
    const int* __restrict__ idxbuf, const float* __restrict__ distbuf,
    float* __restrict__ aggbuf) {
  __shared__ float sW1[8 * 64];       // 7x64 zero-padded to 8x64
  __shared__ float sW2[64 * 64];
  __shared__ float sB1[64];
  __shared__ float sB2[64];
  __shared__ float sGP[4][16 * GPS];  // grouped_points (cols 0-6 = grouped_xyz)
  __shared__ float sA1[4][16 * GPS];  // relu(layer1) activations
  __shared__ int sIdx[4][16];

  const int tid = threadIdx.x;
#if USE_TDM
  if (tid < 32) {                     // wave 0 drives the DMA engine
    tdm_load_f32(w1, &sW1[0], 7 * 64);
    tdm_load_f32(w2, &sW2[0], 64 * 64);
  }
  for (int t = tid + 7 * 64; t < 8 * 64; t += 128) sW1[t] = 0.0f;  // pad row 7
#else
  for (int t = tid; t < 8 * 64; t += 128) {
    int r = t >> 6, c = t & 63;
    sW1[t] = (r < 7) ? w1[r * 64 + c] : 0.0f;
  }
  for (int t = tid; t < 64 * 64; t += 128) sW2[t] = w2[t];
#endif
  if (tid < 64) { sB1[tid] = b1[tid]; sB2[tid] = b2[tid]; }

  const int wave = tid >> 5, lane = tid & 31;
  const int p = blockIdx.x * 4 + wave;           // global point id
  const int b = p >> 12, n = p & (NPTS - 1);
  const size_t nb = (size_t)p * KNN;
  const float* xb = xyz + (size_t)b * NPTS * 3;
  const float* pb = points + (size_t)b * NPTS * CPF;
  const float cx = xb[n * 3 + 0], cy = xb[n * 3 + 1], cz = xb[n * 3 + 2];

  if (lane < 16) {
    const int gi = idxbuf[nb + lane];
    sIdx[wave][lane] = gi;
    const float gx = xb[gi * 3 + 0], gy = xb[gi * 3 + 1], gz = xb[gi * 3 + 2];
    float* row = &sGP[wave][lane * GPS];
    row[0] = gx; row[1] = gy; row[2] = gz;
    row[3] = gx - cx; row[4] = gy - cy; row[5] = gz - cz;
    row[6] = distbuf[nb + lane];
  }
  __syncthreads();
  for (int t = lane; t < 16 * CPF; t += 32) {
    const int r = t / CPF, c = t - r * CPF;
    sGP[wave][r * GPS + 7 + c] = pb[(size_t)sIdx[wave][r] * CPF + c];
  }
  TDM_WAIT();          // issuing wave: weights landed in LDS before publishing
  __syncthreads();

  const int ml = lane & 15;             // A row / D col
  const int kh = (lane >> 4) << 1;      // K offset for upper half-wave

  // layer 1: relu(gxyz[16x8pad] @ W1[8x64] + b1)
  for (int nt = 0; nt < 4; ++nt) {
    const float bias = sB1[nt * 16 + ml];
    v8f acc;
#pragma unroll
    for (int r = 0; r < 8; ++r) acc[r] = bias;
#pragma unroll
    for (int kc = 0; kc < 2; ++kc) {
      v2f a, bb;
#pragma unroll
      for (int r = 0; r < 2; ++r) {
        const int k = kc * 4 + kh + r;
        a[r] = sGP[wave][ml * GPS + k];       // row 7 garbage * zero weight = 0
        bb[r] = sW1[k * 64 + nt * 16 + ml];
      }
      acc = wmma4(a, bb, acc);
    }
#pragma unroll
    for (int r = 0; r < 8; ++r) {
      const int m = (lane < 16) ? r : (r + 8);
      sA1[wave][m * GPS + nt * 16 + ml] = relu(acc[r]);
    }
  }
  __syncthreads();

  // layer 2 + elementwise * grouped_points + mean over the 16 neighbors
  for (int nt = 0; nt < 4; ++nt) {
    const float bias = sB2[nt * 16 + ml];
    v8f acc;
#pragma unroll
    for (int r = 0; r < 8; ++r) acc[r] = bias;
#pragma unroll
    for (int kc = 0; kc < 16; ++kc) {
      v2f a, bb;
#pragma unroll
      for (int r = 0; r < 2; ++r) {
        const int k = kc * 4 + kh + r;
        a[r] = sA1[wave][ml * GPS + k];
        bb[r] = sW2[k * 64 + nt * 16 + ml];
      }
      acc = wmma4(a, bb, acc);
    }
    float colsum = 0.0f;
#pragma unroll
    for (int r = 0; r < 8; ++r) {
      const int m = (lane < 16) ? r : (r + 8);
      colsum += acc[r] * sGP[wave][m * GPS + nt * 16 + ml];
    }
    colsum += __shfl_xor(colsum, 16, 32);   // join both row-halves of the tile
    if (lane < 16)
      aggbuf[(size_t)p * 64 + nt * 16 + lane] = colsum * (1.0f / 16.0f);
  }
}

// ---------------------------------------------------------------------------
// Kernel 3: h = relu([xyz, agg][32768x67] @ mlp_w1[67x128] + b1)
// ---------------------------------------------------------------------------
__global__ __launch_bounds__(128) void mlp1_kernel(
    const float* __restrict__ xyz, const float* __restrict__ aggbuf,
    const float* __restrict__ w1, const float* __restrict__ b1,
    float* __restrict__ hid) {
  __shared__ float sW[68 * 128];   // K padded 67 -> 68, row 67 zeroed
  __shared__ float sB[128];
  const int tid = threadIdx.x;
#if USE_TDM
  if (tid < 32) tdm_load_f32(w1, &sW[0], 67 * 128);
  for (int t = tid + 67 * 128; t < 68 * 128; t += 128) sW[t] = 0.0f;
#else
  for (int t = tid; t < 68 * 128; t += 128) {
    int r = t >> 7, c = t & 127;
    sW[t] = (r < 67) ? w1[r * 128 + c] : 0.0f;
  }
#endif
  sB[tid] = b1[tid];
  TDM_WAIT();
  __syncthreads();

  const int wave = tid >> 5, lane = tid & 31;
  const int rowbase = (blockIdx.x * 4 + wave) * 16;
  const int ml = lane & 15, kh = (lane >> 4) << 1;
  const int row = rowbase + ml;
  const int b = row >> 12, n = row & (NPTS - 1);

  float areg[34];
#pragma unroll
  for (int kc = 0; kc < 17; ++kc) {
#pragma unroll
    for (int r = 0; r < 2; ++r) {
      const int k = kc * 4 + kh + r;
      float v;
      if (k < 3)        v = xyz[((size_t)b * NPTS + n) * 3 + k];
      else if (k < 67)  v = aggbuf[(size_t)row * 64 + (k - 3)];
      else              v = 0.0f;
      areg[kc * 2 + r] = v;
    }
  }
  for (int nt = 0; nt < 8; ++nt) {
    const float bias = sB[nt * 16 + ml];
    v8f acc;
#pragma unroll
    for (int r = 0; r < 8; ++r) acc[r] = bias;
#pragma unroll
    for (int kc = 0; kc < 17; ++kc) {
      v2f a, bb;
      a[0] = areg[kc * 2 + 0];
      a[1] = areg[kc * 2 + 1];
      bb[0] = sW[(kc * 4 + kh + 0) * 128 + nt * 16 + ml];
      bb[1] = sW[(kc * 4 + kh + 1) * 128 + nt * 16 + ml];
      acc = wmma4(a, bb, acc);
    }
#pragma unroll
    for (int r = 0; r < 8; ++r) {
      const int m = (lane < 16) ? r : (r + 8);
      hid[(size_t)(rowbase + m) * 128 + nt * 16 + ml] = relu(acc[r]);
    }
  }
}

// ---------------------------------------------------------------------------
// Kernel 4: out = relu(hid @ mlp_w2 + b2 + points @ sc_w + sc_b)
// Both GEMMs accumulate into one WMMA tile; fused bias seeds the accumulator.
// ---------------------------------------------------------------------------
__global__ __launch_bounds__(128) void mlp2_kernel(
    const float* __restrict__ hid, const float* __restrict__ points,
    const float* __restrict__ w2, const float* __restrict__ b2,
    const float* __restrict__ scw, const float* __restrict__ scb,
    float* __restrict__ out) {
  __shared__ float sW[128 * 64];
  __shared__ float sS[60 * 64];    // K padded 57 -> 60, rows 57-59 zeroed
  __shared__ float sB[64];
  const int tid = threadIdx.x;
#if USE_TDM
  if (tid < 32) {
    tdm_load_f32(w2, &sW[0], 128 * 64);
    tdm_load_f32(scw, &sS[0], 57 * 64);
  }
  for (int t = tid + 57 * 64; t < 60 * 64; t += 128) sS[t] = 0.0f;
#else
  for (int t = tid; t < 128 * 64; t += 128) sW[t] = w2[t];
  for (int t = tid; t < 60 * 64; t += 128) {
    int r = t >> 6, c = t & 63;
    sS[t] = (r < 57) ? scw[r * 64 + c] : 0.0f;
  }
#endif
  if (tid < 64) sB[tid] = b2[tid] + scb[tid];
  TDM_WAIT();
  __syncthreads();

  const int wave = tid >> 5, lane = tid & 31;
  const int rowbase = (blockIdx.x * 4 + wave) * 16;
  const int ml = lane & 15, kh = (lane >> 4) << 1;
  const int row = rowbase + ml;
  const int b = row >> 12, n = row & (NPTS - 1);

  float ah[64];
#pragma unroll
  for (int kc = 0; kc < 32; ++kc)
#pragma unroll
    for (int r = 0; r < 2; ++r)
      ah[kc * 2 + r] = hid[(size_t)row * 128 + kc * 4 + kh + r];

  float ap[30];
#pragma unroll
  for (int kc = 0; kc < 15; ++kc)
#pragma unroll
    for (int r = 0; r < 2; ++r) {
      const int k = kc * 4 + kh + r;
      ap[kc * 2 + r] =
          (k < 57) ? points[((size_t)b * NPTS + n) * CPF + k] : 0.0f;
    }

  for (int nt = 0; nt < 4; ++nt) {
    const float bias = sB[nt * 16 + ml];
    v8f acc;
#pragma unroll
    for (int r = 0; r < 8; ++r) acc[r] = bias;
#pragma unroll
    for (int kc = 0; kc < 32; ++kc) {           // hid @ W2
      v2f a, bb;
      a[0] = ah[kc * 2 + 0];
      a[1] = ah[kc * 2 + 1];
      bb[0] = sW[(kc * 4 + kh + 0) * 64 + nt * 16 + ml];
      bb[1] = sW[(kc * 4 + kh + 1) * 64 + nt * 16 + ml];
      acc = wmma4(a, bb, acc);
    }
#pragma unroll
    for (int kc = 0; kc < 15; ++kc) {           // points @ sc_w (shortcut)
      v2f a, bb;
      a[0] = ap[kc * 2 + 0];
      a[1] = ap[kc * 2 + 1];
      bb[0] = sS[(kc * 4 + kh + 0) * 64 + nt * 16 + ml];
      bb[1] = sS[(kc * 4 + kh + 1) * 64 + nt * 16 + ml];
      acc = wmma4(a, bb, acc);
    }
#pragma unroll
    for (int r = 0; r < 8; ++r) {
      const int m = (lane < 16) ? r : (r + 8);
      out[(size_t)(rowbase + m) * 64 + nt * 16 + ml] = relu(acc[r]);
    }
  }
}

// ---------------------------------------------------------------------------
extern "C" void kernel_launch(void* const* d_in, const int* in_sizes, int n_in,
                              void* d_out, int out_size, void* d_ws,
                              size_t ws_size, hipStream_t stream) {
  (void)in_sizes; (void)n_in; (void)out_size; (void)ws_size;
  const float* xyz    = (const float*)d_in[0];
  const float* points = (const float*)d_in[1];
  const float* dfg_w1 = (const float*)d_in[2];
  const float* dfg_b1 = (const float*)d_in[3];
  const float* dfg_w2 = (const float*)d_in[4];
  const float* dfg_b2 = (const float*)d_in[5];
  const float* mlp_w1 = (const float*)d_in[6];
  const float* mlp_b1 = (const float*)d_in[7];
  const float* mlp_w2 = (const float*)d_in[8];
  const float* mlp_b2 = (const float*)d_in[9];
  const float* sc_w   = (const float*)d_in[10];
  const float* sc_b   = (const float*)d_in[11];
  float* out = (float*)d_out;

  char* ws = (char*)d_ws;
  int*   idxbuf  = (int*)(ws);                             // 2 MB
  float* distbuf = (float*)(ws + ((size_t)2 << 20));       // 2 MB
  float* aggbuf  = (float*)(ws + ((size_t)4 << 20));       // 8 MB
  float* hidbuf  = (float*)(ws + ((size_t)12 << 20));      // 16 MB

  knn_kernel<<<BATCH * NPTS / 256, 256, 0, stream>>>(xyz, idxbuf, distbuf);
  dfg_kernel<<<BATCH * NPTS / 4, 128, 0, stream>>>(
      xyz, points, dfg_w1, dfg_b1, dfg_w2, dfg_b2, idxbuf, distbuf, aggbuf);
  mlp1_kernel<<<BATCH * NPTS / 64, 128, 0, stream>>>(xyz, aggbuf, mlp_w1,
                                                     mlp_b1, hidbuf);
  mlp2_kernel<<<BATCH * NPTS / 64, 128, 0, stream>>>(hidbuf, points, mlp_w2,
                                                     mlp_b2, sc_w, sc_b, out);
}